// ContentAttention_2422361555839
// MI455X (gfx1250) — compile-verified
//
#include <hip/hip_runtime.h>

#define DIM 128
#define NU  8192
#define NI  8192
#define KT  64      // item tile (K tile of the alpha @ Wi GEMM)
#define WP  144     // padded LDS row stride for Wi tile (bank-conflict free)
#define AP  68      // padded LDS row stride for alpha tile

typedef float v2f __attribute__((ext_vector_type(2)));
typedef float v8f __attribute__((ext_vector_type(8)));

// ---------------- kernel 1: w1/w2 = W^T @ a halves (128x128, one block) ----
__global__ void prep_wvec(const float* __restrict__ W, const float* __restrict__ a,
                          float* __restrict__ w1, float* __restrict__ w2) {
  int k = threadIdx.x;
  float s1 = 0.f, s2 = 0.f;
  for (int j = 0; j < DIM; ++j) {
    float wj = W[j * DIM + k];
    s1 += wj * a[j];
    s2 += wj * a[DIM + j];
  }
  w1[k] = s1;
  w2[k] = s2;
}

// ---------------- kernel 2: a1[u]=h_u[u].w1 ; a2[i]=h_i[i].w2 ---------------
__global__ void rowdot(const float* __restrict__ h_u, const float* __restrict__ h_i,
                       const float* __restrict__ w1, const float* __restrict__ w2,
                       float* __restrict__ a1, float* __restrict__ a2) {
  int id = blockIdx.x * blockDim.x + threadIdx.x;
  const float4* rv;
  const float4* wv;
  float* dst;
  if (id < NU) {
    rv = (const float4*)(h_u + (size_t)id * DIM);
    wv = (const float4*)w1;
    dst = a1 + id;
  } else {
    int r = id - NU;
    rv = (const float4*)(h_i + (size_t)r * DIM);
    wv = (const float4*)w2;
    dst = a2 + r;
  }
  float s = 0.f;
#pragma unroll
  for (int t = 0; t < DIM / 4; ++t) {
    float4 x = rv[t], y = wv[t];
    s += x.x * y.x + x.y * y.y + x.z * y.z + x.w * y.w;
  }
  *dst = s;
}

// ---------------- kernel 3: Wi = h_i @ W^T  (row-major [i][r]) --------------
__global__ void compute_wi(const float* __restrict__ h_i, const float* __restrict__ W,
                           float* __restrict__ Wi) {
  int g = blockIdx.x * 256 + threadIdx.x;
  int i = g >> 7;
  int r = g & 127;
  const float4* hr = (const float4*)(h_i + (size_t)i * DIM);
  const float4* wr = (const float4*)(W + (size_t)r * DIM);
  float s = 0.f;
#pragma unroll
  for (int t = 0; t < DIM / 4; ++t) {
    float4 x = hr[t], y = wr[t];
    s += x.x * y.x + x.y * y.y + x.z * y.z + x.w * y.w;
  }
  Wi[g] = s;
}

// ---------------- kernel 4: fused masked softmax + alpha@Wi -----------------
// One workgroup = 16 user rows. 256 threads = 8 wave32.
__global__ __launch_bounds__(256) void attn_main(
    const int* __restrict__ adj, const float* __restrict__ a1,
    const float* __restrict__ a2, const float* __restrict__ Wi,
    float* __restrict__ outp, float* __restrict__ alphap) {
  __shared__ float sWi[KT * WP];   // 36864 B
  __shared__ float sAl[16 * AP];   //  4352 B
  __shared__ float sM[16];
  __shared__ float sInv[16];

  const int tid  = threadIdx.x;
  const int lane = tid & 31;
  const int w    = tid >> 5;       // wave id 0..7
  const int u0   = blockIdx.x * 16;

  // ---- phase 1: per-row online max / sum-exp / neighbor count ----
  for (int rr = 0; rr < 2; ++rr) {
    const int row = w + 8 * rr;
    const float a1v = a1[u0 + row];
    const int* arow = adj + (size_t)(u0 + row) * NI;
    float m = -1e30f, s = 0.f;
    int cnt = 0;
    for (int k = 0; k < NI / 32; ++k) {
      int i = lane + 32 * k;
      int ad = arow[i];
      float t = a1v + a2[i];
      float e = t > 0.f ? t : 0.2f * t;
      if (ad != 0) {
        cnt++;
        if (e > m) { s = s * __expf(m - e) + 1.f; m = e; }
        else       { s += __expf(e - m); }
      }
    }
    // wave32 butterfly combine of (m, s, cnt)
    for (int off = 16; off >= 1; off >>= 1) {
      float mo = __shfl_xor(m, off, 32);
      float so = __shfl_xor(s, off, 32);
      int   co = __shfl_xor(cnt, off, 32);
      float nm = fmaxf(m, mo);
      s = s * __expf(m - nm) + so * __expf(mo - nm);
      m = nm;
      cnt += co;
    }
    if (lane == 0) {
      sM[row]   = m;
      sInv[row] = (cnt > 0) ? (1.f / s) : 0.f;   // no_item rows -> alpha/out = 0
    }
  }
  __syncthreads();

  // ---- phase 2: stream item tiles; build alpha, store it, WMMA into out ----
  const int rA    = lane & 15;   // M row (A) / N col (B)
  const int khalf = lane >> 4;   // K half select
  const int n0    = w * 16;      // this wave's 16 output dims
  v8f acc = {0.f, 0.f, 0.f, 0.f, 0.f, 0.f, 0.f, 0.f};

  for (int it0 = 0; it0 < NI; it0 += KT) {
    // stage Wi tile [KT x 128] into padded LDS
    {
      const float4* src = (const float4*)(Wi + (size_t)it0 * DIM);
#pragma unroll
      for (int j = 0; j < (KT * DIM / 4) / 256; ++j) {   // 8 float4 per thread
        int q    = tid + j * 256;
        int rowi = q >> 5;          // 32 float4 per 128-wide row
        int colv = q & 31;
        float4 v = src[rowi * 32 + colv];
        float* dst = &sWi[rowi * WP + colv * 4];
        dst[0] = v.x; dst[1] = v.y; dst[2] = v.z; dst[3] = v.w;
      }
    }
    // build alpha tile [16 x KT]; nontemporal-store alpha to global
#pragma unroll
    for (int j = 0; j < 4; ++j) {
      int e2 = tid + j * 256;
      int r  = e2 >> 6;
      int i  = e2 & 63;
      int gi = it0 + i;
      int ad = adj[(size_t)(u0 + r) * NI + gi];
      float t  = a1[u0 + r] + a2[gi];
      float ee = t > 0.f ? t : 0.2f * t;
      float al = (ad != 0) ? __expf(ee - sM[r]) * sInv[r] : 0.f;
      sAl[r * AP + i] = al;
      __builtin_nontemporal_store(al, alphap + (size_t)(u0 + r) * NI + gi);
    }
    __syncthreads();

    // 16 steps of v_wmma_f32_16x16x4_f32 over this K tile
#pragma unroll
    for (int ks = 0; ks < KT / 4; ++ks) {
      int kk = ks * 4;
      v2f af, bf;
      // A (alpha 16x4): VGPR0 = K {0|2}, VGPR1 = K {1|3}, lanes = M
      af.x = sAl[rA * AP + kk + 2 * khalf];
      af.y = sAl[rA * AP + kk + 2 * khalf + 1];
      // B (Wi 4x16): VGPR0 = K {0|2}, VGPR1 = K {1|3}, lanes = N
      bf.x = sWi[(kk + 2 * khalf) * WP + n0 + rA];
      bf.y = sWi[(kk + 2 * khalf + 1) * WP + n0 + rA];
      acc = __builtin_amdgcn_wmma_f32_16x16x4_f32(
          /*neg_a=*/false, af, /*neg_b=*/false, bf,
          /*c_mod=*/(short)0, acc, /*reuse_a=*/false, /*reuse_b=*/false);
    }
    __syncthreads();
  }

  // write out tile: VGPR j -> M = j + 8*khalf, N = n0 + rA
#pragma unroll
  for (int j = 0; j < 8; ++j) {
    int M = j + 8 * khalf;
    outp[(size_t)(u0 + M) * DIM + n0 + rA] = acc[j];
  }
}

extern "C" void kernel_launch(void* const* d_in, const int* in_sizes, int n_in,
                              void* d_out, int out_size, void* d_ws, size_t ws_size,
                              hipStream_t stream) {
  const float* h_u = (const float*)d_in[0];
  const float* h_i = (const float*)d_in[1];
  const int*   adj = (const int*)d_in[2];
  const float* W   = (const float*)d_in[3];
  const float* a   = (const float*)d_in[4];

  float* ws = (float*)d_ws;
  float* w1 = ws;                 // 128
  float* w2 = ws + 128;           // 128
  float* a1 = ws + 256;           // 8192
  float* a2 = ws + 8448;          // 8192
  float* Wi = ws + 16640;         // 8192*128

  float* outp   = (float*)d_out;                     // [8192 x 128]
  float* alphap = outp + (size_t)NU * DIM;           // [8192 x 8192]

  prep_wvec<<<1, DIM, 0, stream>>>(W, a, w1, w2);
  rowdot<<<(NU + NI) / 256, 256, 0, stream>>>(h_u, h_i, w1, w2, a1, a2);
  compute_wi<<<(NI * DIM) / 256, 256, 0, stream>>>(h_i, W, Wi);
  attn_main<<<NU / 16, 256, 0, stream>>>(adj, a1, a2, Wi, outp, alphap);
}